// GCN_82085414961196
// MI455X (gfx1250) — compile-verified
//
#include <hip/hip_runtime.h>
#include <cstdint>

#define HID 128
#define NGRAPH 64
#define NEG_SLOPE 0.01f

typedef float v2f __attribute__((ext_vector_type(2)));
typedef float v8f __attribute__((ext_vector_type(8)));

// LDS row pitches (floats), padded to avoid bank conflicts:
//  As read: lanes 0-15 stride ASP -> banks 4*i (distinct)
//  Bs read: half-waves offset by 2 rows -> 2*BSP = 144 ≡ 16 (mod 64) -> disjoint bank sets
#define ASP 132
#define BSP 72

// ---------------- CDNA5 async global->LDS copy (ASYNCcnt-tracked) ----------------
__device__ __forceinline__ void async_copy_b128(uint32_t lds_off, const float* gptr) {
    asm volatile("global_load_async_to_lds_b128 %0, %1, off"
                 :: "v"(lds_off), "v"((uint64_t)(uintptr_t)gptr)
                 : "memory");
}
__device__ __forceinline__ void wait_asynccnt0() {
    asm volatile("s_wait_asynccnt 0x0" ::: "memory");
}

// ---------------- elementwise helpers ----------------
__global__ void k_fill(float* __restrict__ p, int n, float v) {
    int i = blockIdx.x * blockDim.x + threadIdx.x;
    if (i < n) p[i] = v;
}

__global__ void k_leaky(float* __restrict__ p, int n) {
    int i = blockIdx.x * blockDim.x + threadIdx.x;
    if (i < n) { float v = p[i]; p[i] = v > 0.f ? v : NEG_SLOPE * v; }
}

// ---------------- degree / normalization ----------------
__global__ void k_deg(const int* __restrict__ ei, const float* __restrict__ w,
                      float* __restrict__ deg, int E) {
    int e = blockIdx.x * blockDim.x + threadIdx.x;
    if (e < E) atomicAdd(&deg[ei[E + e]], w[e]);   // col = ei[1][e]
}

__global__ void k_rsqrt(float* __restrict__ p, int n) {
    int i = blockIdx.x * blockDim.x + threadIdx.x;
    if (i < n) { float d = p[i]; p[i] = d > 0.f ? rsqrtf(d) : 0.f; }
}

// ---------------- fp32 WMMA GEMM: C[M,Nc] = A[M,K] * B[K,Nc] ----------------
// Block = 128 threads = 4 waves. All 4 waves share one 16-row M tile; the block
// covers 4 consecutive 16-col N tiles. A tile (16xK) and B slab (Kx64) are
// async-copied into LDS once per block; WMMA fragments then come from LDS.
// Requires: M % 16 == 0, K % 4 == 0, Nc % 64 == 0 (here M=40000, K=Nc=128).
__global__ void __launch_bounds__(128)
k_gemm_wmma(const float* __restrict__ A, const float* __restrict__ B,
            float* __restrict__ C, int M, int K, int Nc) {
    __shared__ float As[16 * ASP];     //  8448 B
    __shared__ float Bs[HID * BSP];    // 36864 B

    int blocksPerRow = Nc >> 6;                  // N tiles / 4
    int tm  = blockIdx.x / blocksPerRow;
    int tn0 = (blockIdx.x - tm * blocksPerRow) << 2;
    int tid = threadIdx.x;

    uint32_t asBase = (uint32_t)(uintptr_t)(void*)As;  // low 32 bits = LDS offset
    uint32_t bsBase = (uint32_t)(uintptr_t)(void*)Bs;

    // ---- stage A: 16 rows x K floats, 16 B per lane per issue ----
    // (K/4 float4 per row; 16*K/4 transfers over 128 threads)
    int aXfers = (16 * K) >> 2;                  // float4 count (= 512 for K=128)
    for (int idx = tid; idx < aXfers; idx += 128) {
        int r  = idx / (K >> 2);
        int c4 = idx - r * (K >> 2);
        async_copy_b128(asBase + (uint32_t)(r * ASP + c4 * 4) * 4u,
                        A + (size_t)(tm * 16 + r) * K + c4 * 4);
    }
    // ---- stage B slab: K rows x 64 cols ----
    int bXfers = (K * 64) >> 2;                  // (= 2048 for K=128)
    for (int idx = tid; idx < bXfers; idx += 128) {
        int r  = idx >> 4;                       // 16 float4 per row
        int c4 = idx & 15;
        async_copy_b128(bsBase + (uint32_t)(r * BSP + c4 * 4) * 4u,
                        B + (size_t)r * Nc + (tn0 << 4) + c4 * 4);
    }
    wait_asynccnt0();
    __syncthreads();

    int lane = tid & 31;
    int half = lane >> 4;          // 0: lanes 0-15, 1: lanes 16-31
    int l15  = lane & 15;
    int waveId = tid >> 5;
    int colL = (waveId << 4) + l15;              // column within the 64-wide slab

    v8f acc = {};
    for (int k = 0; k < K; k += 4) {
        int ka = k + (half << 1);                // lanes 16-31 carry K+2,K+3
        v2f a; a.x = As[l15 * ASP + ka];   a.y = As[l15 * ASP + ka + 1];
        v2f b; b.x = Bs[ka * BSP + colL];  b.y = Bs[(ka + 1) * BSP + colL];
        acc = __builtin_amdgcn_wmma_f32_16x16x4_f32(
                  false, a, false, b, (short)0, acc, false, false);
    }
    // D layout: VGPR r -> row = tm*16 + r + half*8, col = (tn0+waveId)*16 + l15
    int col = ((tn0 + waveId) << 4) + l15;
#pragma unroll
    for (int r = 0; r < 8; ++r) {
        C[(size_t)(tm * 16 + r + half * 8) * Nc + col] = acc[r];
    }
}

// ---------------- out = dinv^2 * t + bias  (self-loop term) ----------------
__global__ void k_init_out(float* __restrict__ out, const float* __restrict__ t,
                           const float* __restrict__ dinv, const float* __restrict__ b,
                           int total) {
    int i = blockIdx.x * blockDim.x + threadIdx.x;
    if (i >= total) return;
    int node = i >> 7;                // /HID
    float di = dinv[node];
    out[i] = di * di * t[i] + b[i & (HID - 1)];
}

// ---------------- per-edge scatter: one wave per edge ----------------
__global__ void k_edge_scatter(const float* __restrict__ t, const int* __restrict__ ei,
                               const float* __restrict__ w, const float* __restrict__ dinv,
                               float* __restrict__ out, int E) {
    int gw = (blockIdx.x * blockDim.x + threadIdx.x) >> 5;
    if (gw >= E) return;
    int lane = threadIdx.x & 31;
    int r = ei[gw];
    int c = ei[E + gw];
    float nrm = dinv[r] * w[gw] * dinv[c];
    float4 v = ((const float4*)(t + (size_t)r * HID))[lane];   // 32 lanes x 4 = 128 feats
    float* o = out + (size_t)c * HID + lane * 4;
    atomicAdd(o + 0, nrm * v.x);
    atomicAdd(o + 1, nrm * v.y);
    atomicAdd(o + 2, nrm * v.z);
    atomicAdd(o + 3, nrm * v.w);
}

// ---------------- mean pooling ----------------
__global__ void k_cnt(const int* __restrict__ batch, float* __restrict__ cnt, int n) {
    int i = blockIdx.x * blockDim.x + threadIdx.x;
    if (i < n) atomicAdd(&cnt[batch[i]], 1.0f);
}

__global__ void k_pool(const float* __restrict__ h, const int* __restrict__ batch,
                       float* __restrict__ pool, int total) {
    int i = blockIdx.x * blockDim.x + threadIdx.x;
    if (i >= total) return;
    int node = i >> 7;
    atomicAdd(&pool[batch[node] * HID + (i & (HID - 1))], h[i]);
}

__global__ void k_mean(float* __restrict__ pool, const float* __restrict__ cnt, int total) {
    int i = blockIdx.x * blockDim.x + threadIdx.x;
    if (i < total) pool[i] /= fmaxf(cnt[i >> 7], 1.0f);
}

// ---------------- tiny dense MLP layer ----------------
__global__ void k_mlp(const float* __restrict__ in, const float* __restrict__ Wm,
                      const float* __restrict__ b, float* __restrict__ out,
                      int G, int K, int N, int act) {
    int i = blockIdx.x * blockDim.x + threadIdx.x;
    if (i >= G * N) return;
    int g = i / N, n = i - g * N;
    float s = b[n];
    for (int k = 0; k < K; ++k) s += in[g * K + k] * Wm[k * N + n];
    if (act) s = s > 0.f ? s : NEG_SLOPE * s;
    out[i] = s;
}

extern "C" void kernel_launch(void* const* d_in, const int* in_sizes, int n_in,
                              void* d_out, int out_size, void* d_ws, size_t ws_size,
                              hipStream_t stream) {
    const float* x   = (const float*)d_in[0];
    const int*   ei  = (const int*)d_in[1];     // [2, E] flat
    const float* ew  = (const float*)d_in[2];
    const int*   bat = (const int*)d_in[3];
    const float* W[3]  = { (const float*)d_in[4], (const float*)d_in[6], (const float*)d_in[8] };
    const float* bb[3] = { (const float*)d_in[5], (const float*)d_in[7], (const float*)d_in[9] };
    const float* fw1 = (const float*)d_in[10]; const float* fb1 = (const float*)d_in[11];
    const float* fw2 = (const float*)d_in[12]; const float* fb2 = (const float*)d_in[13];
    const float* fw3 = (const float*)d_in[14]; const float* fb3 = (const float*)d_in[15];
    float* out = (float*)d_out;

    const int E = in_sizes[2];          // edge count
    const int N = in_sizes[3];          // node count
    const int G = NGRAPH;

    float* ws   = (float*)d_ws;
    float* t    = ws;                          // [N, HID] GEMM result
    float* h    = t + (size_t)N * HID;         // [N, HID] layer output
    float* dinv = h + (size_t)N * HID;         // [N]
    float* pool = dinv + N;                    // [G, HID] (+cnt contiguous after)
    float* cnt  = pool + (size_t)G * HID;      // [G]
    float* g1   = cnt + G;                     // [G, 64]
    float* g2   = g1 + (size_t)G * 64;         // [G, 32]

    const int NH = N * HID;

    // ---- normalization: deg = 1 (self loop) + sum_w at col; dinv = rsqrt(deg) ----
    k_fill<<<(N + 255) / 256, 256, 0, stream>>>(dinv, N, 1.0f);
    k_fill<<<(G * HID + G + 255) / 256, 256, 0, stream>>>(pool, G * HID + G, 0.0f);
    k_deg<<<(E + 255) / 256, 256, 0, stream>>>(ei, ew, dinv, E);
    k_rsqrt<<<(N + 255) / 256, 256, 0, stream>>>(dinv, N);

    // ---- 3 GCN layers ----
    const float* in = x;
    int gemmBlocks = (N / 16) * (HID / 64);                  // M tiles * (N tiles / 4)
    long long scatterThreads = (long long)E * 32;
    int scatterBlocks = (int)((scatterThreads + 255) / 256);

    for (int L = 0; L < 3; ++L) {
        k_gemm_wmma<<<gemmBlocks, 128, 0, stream>>>(in, W[L], t, N, HID, HID);
        k_init_out<<<(NH + 255) / 256, 256, 0, stream>>>(h, t, dinv, bb[L], NH);
        k_edge_scatter<<<scatterBlocks, 256, 0, stream>>>(t, ei, ew, dinv, h, E);
        if (L < 2) k_leaky<<<(NH + 255) / 256, 256, 0, stream>>>(h, NH);
        in = h;
    }

    // ---- global mean pool ----
    k_cnt<<<(N + 255) / 256, 256, 0, stream>>>(bat, cnt, N);
    k_pool<<<(NH + 255) / 256, 256, 0, stream>>>(h, bat, pool, NH);
    k_mean<<<(G * HID + 255) / 256, 256, 0, stream>>>(pool, cnt, G * HID);

    // ---- MLP head ----
    k_mlp<<<(G * 64 + 255) / 256, 256, 0, stream>>>(pool, fw1, fb1, g1, G, 128, 64, 1);
    k_mlp<<<(G * 32 + 255) / 256, 256, 0, stream>>>(g1, fw2, fb2, g2, G, 64, 32, 1);
    k_mlp<<<(G * 2 + 255) / 256, 256, 0, stream>>>(g2, fw3, fb3, out, G, 32, 2, 0);
}